// TextDetectionLoss_69861938037475
// MI455X (gfx1250) — compile-verified
//
#include <hip/hip_runtime.h>
#include <math.h>

// Problem constants (match reference setup_inputs)
#define BB 64
#define NN 32
#define HH 320
#define WW 320
#define NE (BB * HH * WW)   // 6,553,600 elements in text map
#define NF4 (NE / 4)        // 1,638,400 float4s
#define MM (NN * 5)         // 160 sample points per batch image
#define EPSF 1e-6f

typedef __attribute__((ext_vector_type(2))) float v2f;
typedef __attribute__((ext_vector_type(4))) float v4f;
typedef __attribute__((ext_vector_type(8))) float v8f;

// Workspace layout (floats):
// [0]=sum BCE, [1]=sum p*t, [2]=sum p, [3]=sum t,
// [4]=sum -log(iou+eps), [5]=sum (1-giou), [6]=sum smoothL1, [7]=sum conf BCE

__global__ void tdl_zero_ws(float* __restrict__ gsum) {
    if (threadIdx.x < 8) gsum[threadIdx.x] = 0.0f;
}

// ---------------------------------------------------------------------------
// Kernel 1: streaming BCE + Dice partial sums over 6.55M elements.
// Bandwidth-bound: 52.4 MB read once -> ~2.3 us floor at 23.3 TB/s.
// Wave-level reduction finishes with V_WMMA_F32_16X16X4_F32:
//   fold 32 lanes -> 4 groups (shfl_xor 4,2,1), place group sums as the
//   B matrix (4x16: k = lane-group, n = quantity), A = ones(16x4),
//   D[0][q] = sum_k B[k][q] = wave total for quantity q (lives in lane q).
// ---------------------------------------------------------------------------
__global__ void tdl_map_reduce(const float* __restrict__ pred,
                               const float* __restrict__ tgt,
                               float* __restrict__ gsum) {
    const int tid    = threadIdx.x;
    const int gid    = blockIdx.x * blockDim.x + tid;
    const int stride = gridDim.x * blockDim.x;

    float s_bce = 0.0f, s_pt = 0.0f, s_p = 0.0f, s_t = 0.0f;

    const v4f* p4 = (const v4f*)pred;
    const v4f* t4 = (const v4f*)tgt;

    // NF4 is an exact multiple of the grid size (640*256*10), so every lane
    // executes the same trip count -> EXEC stays all-ones into the WMMA.
    for (int i = gid; i < NF4; i += stride) {
        v4f p = __builtin_nontemporal_load(p4 + i);  // one-shot stream: NT hint
        v4f t = __builtin_nontemporal_load(t4 + i);
#pragma unroll
        for (int j = 0; j < 4; ++j) {
            float pv  = p[j];
            float tv  = t[j];
            float lp  = fmaxf(logf(pv), -100.0f);
            float l1p = fmaxf(log1pf(-pv), -100.0f);
            s_bce -= tv * lp + (1.0f - tv) * l1p;
            s_pt  += pv * tv;
            s_p   += pv;
            s_t   += tv;
        }
    }

    // ---- fold 32 lanes into 4 groups of 8 (bits 2:0 collapsed) ----
    s_bce += __shfl_xor(s_bce, 4); s_pt += __shfl_xor(s_pt, 4);
    s_p   += __shfl_xor(s_p,   4); s_t  += __shfl_xor(s_t,  4);
    s_bce += __shfl_xor(s_bce, 2); s_pt += __shfl_xor(s_pt, 2);
    s_p   += __shfl_xor(s_p,   2); s_t  += __shfl_xor(s_t,  2);
    s_bce += __shfl_xor(s_bce, 1); s_pt += __shfl_xor(s_pt, 1);
    s_p   += __shfl_xor(s_p,   1); s_t  += __shfl_xor(s_t,  1);
    // lane L now holds G[L>>3][q] for all q (group sum of its 8-lane group)

    const int lane = tid & 31;
    const int q    = lane & 3;
    // select this lane's "column quantity" value
    float v = (q == 0) ? s_bce : (q == 1) ? s_pt : (q == 2) ? s_p : s_t;
    // partner group's value for the same quantity-column
    float w = __shfl_xor(v, 8);
    const int hi8 = (lane >> 3) & 1;

    // B matrix (4x16): lanes 0-15 carry rows k=0,1; lanes 16-31 rows k=2,3.
    // Column n = quantity index (only columns 0..3 of D are consumed).
    v2f bm;
    bm[0] = hi8 ? w : v;
    bm[1] = hi8 ? v : w;
    v2f am;
    am[0] = 1.0f;   // A = ones(16x4): D[m][n] = sum_k B[k][n]
    am[1] = 1.0f;
    v8f c = {0.0f, 0.0f, 0.0f, 0.0f, 0.0f, 0.0f, 0.0f, 0.0f};

    // (neg_a, A, neg_b, B, c_mod, C, reuse_a, reuse_b)
    v8f d = __builtin_amdgcn_wmma_f32_16x16x4_f32(
        false, am, false, bm, (short)0, c, false, false);
    // d[0] at lane q == wave-total of quantity q

    __shared__ float sred[4];
    if (tid < 4) sred[tid] = 0.0f;
    __syncthreads();
    if (lane < 4) atomicAdd(&sred[lane], d[0]);
    __syncthreads();
    if (tid < 4) atomicAdd(&gsum[tid], sred[tid]);
}

// ---------------------------------------------------------------------------
// Kernel 2: box loss over B*N = 2048 boxes, 5 sample points each.
// Gathers 5 conf + 20 bbox floats per box; trivially small.
// ---------------------------------------------------------------------------
__global__ void tdl_box(const float* __restrict__ conf_map,
                        const float* __restrict__ bbox,
                        const float* __restrict__ boxes,
                        float* __restrict__ gsum) {
    const int idx = blockIdx.x * blockDim.x + threadIdx.x;
    float s_nli = 0.0f, s_gio = 0.0f, s_reg = 0.0f, s_cf = 0.0f;

    if (idx < BB * NN) {
        const int b = idx / NN;
        const float* bx = boxes + (size_t)idx * 5;
        const float tc = bx[0];
        float nx1 = fminf(fmaxf(bx[1] / (float)WW, 0.0f), 0.999f);
        float ny1 = fminf(fmaxf(bx[2] / (float)HH, 0.0f), 0.999f);
        float nx2 = fmaxf(nx1 + 0.001f, fminf(bx[3] / (float)WW, 1.0f));
        float ny2 = fmaxf(ny1 + 0.001f, fminf(bx[4] / (float)HH, 1.0f));

        int x1p = min(max((int)(nx1 * WW), 0), WW - 1);
        int y1p = min(max((int)(ny1 * HH), 0), HH - 1);
        int x2p = max(x1p + 1, min((int)(nx2 * WW), WW - 1));
        int y2p = max(y1p + 1, min((int)(ny2 * HH), HH - 1));
        int cx = (x1p + x2p) >> 1;
        int cy = (y1p + y2p) >> 1;

        const int xs[5] = {cx, x1p, x2p, x1p, x2p};
        const int ys[5] = {cy, y1p, y1p, y2p, y2p};

        const float ta = (nx2 - nx1) * (ny2 - ny1);
        const float* cmap = conf_map + (size_t)b * HH * WW;
        const float* bmap = bbox + (size_t)b * 4 * HH * WW;

#pragma unroll
        for (int s = 0; s < 5; ++s) {
            const int off = ys[s] * WW + xs[s];
            float g0 = bmap[off];
            float g1 = bmap[HH * WW + off];
            float g2 = bmap[2 * HH * WW + off];
            float g3 = bmap[3 * HH * WW + off];
            float gc = cmap[off];

            float px1 = fminf(fmaxf(g0, 0.0f), 0.99f);
            float py1 = fminf(fmaxf(g1, 0.0f), 0.99f);
            float px2 = fminf(fmaxf(g2, 0.01f), 1.0f);
            float py2 = fminf(fmaxf(g3, 0.01f), 1.0f);
            float cx1 = fminf(px1, px2), cx2 = fmaxf(px1, px2);
            float cy1 = fminf(py1, py2), cy2 = fmaxf(py1, py2);

            float pa    = (cx2 - cx1) * (cy2 - cy1);
            float ix    = fmaxf(fminf(cx2, nx2) - fmaxf(cx1, nx1), 0.0f);
            float iy    = fmaxf(fminf(cy2, ny2) - fmaxf(cy1, ny1), 0.0f);
            float inter = ix * iy;
            float uni   = pa + ta - inter;
            float iou   = inter / (uni + EPSF);
            float ex    = fmaxf(cx2, nx2) - fminf(cx1, nx1);
            float ey    = fmaxf(cy2, ny2) - fminf(cy1, ny1);
            float enc   = ex * ey;
            float giou  = iou - (enc - uni) / (enc + EPSF);

            s_nli += -logf(iou + EPSF);
            s_gio += 1.0f - giou;

            float dv0 = cx1 - nx1, dv1 = cy1 - ny1, dv2 = cx2 - nx2, dv3 = cy2 - ny2;
            float a0 = fabsf(dv0); s_reg += (a0 < 1.0f) ? 0.5f * a0 * a0 : a0 - 0.5f;
            float a1 = fabsf(dv1); s_reg += (a1 < 1.0f) ? 0.5f * a1 * a1 : a1 - 0.5f;
            float a2 = fabsf(dv2); s_reg += (a2 < 1.0f) ? 0.5f * a2 * a2 : a2 - 0.5f;
            float a3 = fabsf(dv3); s_reg += (a3 < 1.0f) ? 0.5f * a3 * a3 : a3 - 0.5f;

            float lp  = fmaxf(logf(gc), -100.0f);
            float l1p = fmaxf(log1pf(-gc), -100.0f);
            s_cf += -(tc * lp + (1.0f - tc) * l1p);
        }
    }

    __shared__ float sr[4];
    if (threadIdx.x < 4) sr[threadIdx.x] = 0.0f;
    __syncthreads();
    atomicAdd(&sr[0], s_nli);
    atomicAdd(&sr[1], s_gio);
    atomicAdd(&sr[2], s_reg);
    atomicAdd(&sr[3], s_cf);
    __syncthreads();
    if (threadIdx.x < 4) atomicAdd(&gsum[4 + threadIdx.x], sr[threadIdx.x]);
}

// ---------------------------------------------------------------------------
// Kernel 3: combine 8 partial sums into (total, text_map_loss, box_loss, conf_loss)
// ---------------------------------------------------------------------------
__global__ void tdl_finalize(const float* __restrict__ gsum, float* __restrict__ out) {
    if (threadIdx.x == 0 && blockIdx.x == 0) {
        float mean_bce = gsum[0] / (float)NE;
        float dice     = (2.0f * gsum[1] + 1e-5f) / (gsum[2] + gsum[3] + 1e-5f);
        float tml      = 0.5f * mean_bce + 0.5f * (1.0f - dice);

        float denom  = (float)(BB * MM);
        float iou_l  = gsum[4] / denom;
        float giou_l = gsum[5] / denom;
        float reg_l  = gsum[6] / (denom * 4.0f);
        float box_l  = 0.5f * iou_l + 0.3f * reg_l + 0.2f * giou_l;
        float conf_l = gsum[7] / denom;

        float total = tml + 20.0f * box_l + 0.5f * conf_l;
        out[0] = total;
        out[1] = tml;
        out[2] = box_l;
        out[3] = conf_l;
    }
}

extern "C" void kernel_launch(void* const* d_in, const int* in_sizes, int n_in,
                              void* d_out, int out_size, void* d_ws, size_t ws_size,
                              hipStream_t stream) {
    const float* pred  = (const float*)d_in[0];  // text_map_pred [B,1,H,W]
    const float* conf  = (const float*)d_in[1];  // confidence    [B,1,H,W]
    const float* bbox  = (const float*)d_in[2];  // bbox_coords   [B,4,H,W]
    const float* tgt   = (const float*)d_in[3];  // text_map_tgt  [B,1,H,W]
    const float* boxes = (const float*)d_in[4];  // boxes         [B,N,5]
    float* out  = (float*)d_out;
    float* gsum = (float*)d_ws;

    hipLaunchKernelGGL(tdl_zero_ws, dim3(1), dim3(32), 0, stream, gsum);
    // 640 blocks * 256 threads: exactly 10 float4 iterations per thread
    hipLaunchKernelGGL(tdl_map_reduce, dim3(640), dim3(256), 0, stream, pred, tgt, gsum);
    hipLaunchKernelGGL(tdl_box, dim3(8), dim3(256), 0, stream, conf, bbox, boxes, gsum);
    hipLaunchKernelGGL(tdl_finalize, dim3(1), dim3(1), 0, stream, gsum, out);
}